// Model_45243185496174
// MI455X (gfx1250) — compile-verified
//
#include <hip/hip_runtime.h>
#include <hip/hip_bf16.h>
#include <stdint.h>

// ---------------------------------------------------------------------------
// DGCNN (4x GCNConv + SortPool(k=30) + conv/conv/fc/fc + log_softmax)
// Strategy: graphs are 100 contiguous nodes; run each graph inside one
// workgroup entirely in LDS. GEMMs use V_WMMA_F32_16X16X4_F32 (full f32).
// Edge aggregation uses ds_add_f32 LDS atomics (no global atomics in the
// hot loop). Edges are binned per graph once per launch.
// ---------------------------------------------------------------------------

typedef __attribute__((ext_vector_type(2))) float v2f;
typedef __attribute__((ext_vector_type(8))) float v8f;

#define N_NODES 100000
#define N_EDGES 3200000
#define N_GRAPHS 1000
#define NPG 100
#define KTOP 30
#define HSTRIDE 100   // padded row stride of concat feature buffer (97 used)

#if defined(__has_builtin)
#  if __has_builtin(__builtin_amdgcn_wmma_f32_16x16x4_f32)
#    define HAVE_WMMA_F32X4 1
#  endif
#endif

// ---------------------------------------------------------------------------
// K0: zero counters
__global__ __launch_bounds__(256) void k_zero(int* deg, int* gcnt, int* gcur) {
    int i = blockIdx.x * 256 + threadIdx.x;
    if (i < N_NODES) deg[i] = 0;
    if (i < N_GRAPHS) { gcnt[i] = 0; gcur[i] = 0; }
}

// K1: per-node in-degree (excluding self loops) + per-graph edge count
__global__ __launch_bounds__(256) void k_deg(const int* __restrict__ ei,
                                             int* deg, int* gcnt) {
    int e = blockIdx.x * 256 + threadIdx.x;
    if (e >= N_EDGES) return;
    int s = ei[e];
    int d = ei[N_EDGES + e];
    if (s != d) {
        atomicAdd(&deg[d], 1);
        atomicAdd(&gcnt[s / NPG], 1);
    }
}

// K2: dis = rsqrt(deg + 1)  (self loop included)
__global__ __launch_bounds__(256) void k_rsqrt(const int* __restrict__ deg,
                                               float* dis) {
    int i = blockIdx.x * 256 + threadIdx.x;
    if (i < N_NODES) dis[i] = rsqrtf((float)deg[i] + 1.0f);
}

// K3: exclusive scan of per-graph edge counts (single 1024-thread block)
__global__ __launch_bounds__(1024) void k_scan(const int* __restrict__ gcnt,
                                               int* goffs) {
    __shared__ int s[1024];
    int i = threadIdx.x;
    int v0 = (i < N_GRAPHS) ? gcnt[i] : 0;
    s[i] = v0;
    __syncthreads();
    for (int off = 1; off < 1024; off <<= 1) {
        int v = (i >= off) ? s[i - off] : 0;
        __syncthreads();
        s[i] += v;
        __syncthreads();
    }
    if (i < N_GRAPHS) goffs[i] = s[i] - v0;   // exclusive
}

// K4: scatter edges into per-graph bins, packed (srcLocal | dstLocal<<8)
__global__ __launch_bounds__(256) void k_scatter(const int* __restrict__ ei,
                                                 const int* __restrict__ goffs,
                                                 int* gcur, uint32_t* binned) {
    int e = blockIdx.x * 256 + threadIdx.x;
    if (e >= N_EDGES) return;
    int s = ei[e];
    int d = ei[N_EDGES + e];
    if (s == d) return;
    int g = s / NPG;
    int pos = goffs[g] + atomicAdd(&gcur[g], 1);
    binned[pos] = (uint32_t)(s - g * NPG) | ((uint32_t)(d - g * NPG) << 8);
}

// ---------------------------------------------------------------------------
// K5: one fused GCN layer, one workgroup (256 thr = 8 waves) per graph.
//   h = X @ W          (WMMA f32 16x16x4, M padded 100->112)
//   agg[d] += dis[s]*dis[d]*h[s]  over graph edges  (LDS atomics)
//   out = tanh(agg + (1/deg)*h + b)  -> concat buffer column block
template <int FIN, int FOUT>
__global__ __launch_bounds__(256) void k_gcn(
    const float* __restrict__ xin, int in_stride, int in_coloff,
    const float* __restrict__ W, const float* __restrict__ bias,
    const float* __restrict__ dis, const uint32_t* __restrict__ binned,
    const int* __restrict__ goffs, const int* __restrict__ gcnt,
    float* __restrict__ hcat, int out_coloff) {

    constexpr int XS = FIN + 4;                   // padded LDS stride (banking)
    constexpr int HS = (FOUT == 1) ? 1 : FOUT + 1;

    __shared__ float Xs[112 * XS];                // input tile (rows >=100 zero)
    __shared__ float Wt[((FOUT == 1) ? 1 : FOUT) * XS]; // W transposed [n][k]
    __shared__ float hbuf[112 * HS];              // X@W
    __shared__ float agg[NPG * HS];
    __shared__ float disL[NPG];

    const int g = blockIdx.x;
    const int tid = threadIdx.x;

    for (int idx = tid; idx < 112 * FIN; idx += 256) {
        int m = idx / FIN, k = idx - m * FIN;
        Xs[m * XS + k] =
            (m < NPG) ? xin[(g * NPG + m) * in_stride + in_coloff + k] : 0.0f;
    }
    for (int idx = tid; idx < FIN * FOUT; idx += 256) {
        int k = idx / FOUT, n = idx - k * FOUT;
        Wt[n * XS + k] = W[idx];                  // W is [FIN][FOUT] row-major
    }
    for (int i = tid; i < NPG; i += 256) disL[i] = dis[g * NPG + i];
    for (int idx = tid; idx < NPG * HS; idx += 256) agg[idx] = 0.0f;
    __syncthreads();

    if constexpr (FOUT > 1) {
#if defined(HAVE_WMMA_F32X4)
        // --- WMMA path: 7 M-tiles x (FOUT/16) N-tiles, K in steps of 4 ---
        const int wv = tid >> 5, lane = tid & 31;
        constexpr int TN = FOUT / 16;
        const int lm = lane & 15;
        const int khalf = (lane >> 4) << 1;       // 0 or 2
        for (int t = wv; t < 7 * TN; t += 8) {
            int tm = t / TN, tn = t - tm * TN;
            v8f acc = {0.f, 0.f, 0.f, 0.f, 0.f, 0.f, 0.f, 0.f};
            const float* arow = &Xs[(tm * 16 + lm) * XS];
            const float* brow = &Wt[(tn * 16 + lm) * XS];
            for (int k0 = 0; k0 < FIN; k0 += 4) {
                v2f a = *(const v2f*)(arow + k0 + khalf);
                v2f b = *(const v2f*)(brow + k0 + khalf);
                acc = __builtin_amdgcn_wmma_f32_16x16x4_f32(
                    false, a, false, b, (short)0, acc, false, false);
            }
            int rbase = tm * 16 + ((lane >> 4) << 3);
            int col = tn * 16 + lm;
#pragma unroll
            for (int v = 0; v < 8; v++)
                hbuf[(rbase + v) * HS + col] = acc[v];
        }
#else
        // Scalar fallback (keeps compile alive if builtin is absent)
        for (int idx = tid; idx < 112 * FOUT; idx += 256) {
            int m = idx / FOUT, n = idx - m * FOUT;
            float acc = 0.0f;
            for (int k = 0; k < FIN; k++) acc += Xs[m * XS + k] * Wt[n * XS + k];
            hbuf[m * HS + n] = acc;
        }
#endif
    } else {
        // FOUT == 1: plain dot products
        for (int m = tid; m < 112; m += 256) {
            float acc = 0.0f;
#pragma unroll
            for (int k = 0; k < FIN; k++) acc += Xs[m * XS + k] * Wt[k];
            hbuf[m] = acc;
        }
    }
    __syncthreads();

    // --- edge aggregation in LDS ---
    const int base = goffs[g];
    const int ec = gcnt[g];
    for (int e = tid; e < ec; e += 256) {
        uint32_t p = binned[base + e];
        int s = (int)(p & 255u);
        int d = (int)((p >> 8) & 255u);
        float nm = disL[s] * disL[d];
#pragma unroll
        for (int c = 0; c < FOUT; c++)
            atomicAdd(&agg[d * HS + c], nm * hbuf[s * HS + c]);
    }
    __syncthreads();

    // --- combine + tanh + write to concat buffer ---
    for (int idx = tid; idx < NPG * FOUT; idx += 256) {
        int m = idx / FOUT, c = idx - m * FOUT;
        float sn = disL[m] * disL[m];
        float v = agg[m * HS + c] + sn * hbuf[m * HS + c] + bias[c];
        hcat[(g * NPG + m) * HSTRIDE + out_coloff + c] = tanhf(v);
    }
}

// ---------------------------------------------------------------------------
// K6: fused head per graph: stable top-30 sort by channel 96, conv5(relu),
// maxpool2, conv6(relu), fc1(relu), fc2, log_softmax.
__global__ __launch_bounds__(256) void k_head(
    const float* __restrict__ hcat,
    const float* __restrict__ w5, const float* __restrict__ b5,
    const float* __restrict__ w6, const float* __restrict__ b6,
    const float* __restrict__ f1w, const float* __restrict__ f1b,
    const float* __restrict__ f2w, const float* __restrict__ f2b,
    float* __restrict__ out) {

    __shared__ float vals[NPG];
    __shared__ int sel[KTOP];
    __shared__ float pooled[KTOP * 98];   // [30][97], stride 98
    __shared__ float A5[KTOP * 16];       // [30][16]
    __shared__ float P[15 * 16];          // [15][16]
    __shared__ float Z[352];              // conv6 out, flat co*11+t
    __shared__ float O1[128];
    __shared__ float O2[10];
    __shared__ float red[2];

    const int g = blockIdx.x;
    const int tid = threadIdx.x;

    // sort key = last channel (col 96)
    for (int i = tid; i < NPG; i += 256)
        vals[i] = hcat[(g * NPG + i) * HSTRIDE + 96];
    __syncthreads();

    // stable descending rank (argsort(-v) semantics: ties -> lower index first)
    if (tid < NPG) {
        float vi = vals[tid];
        int r = 0;
        for (int j = 0; j < NPG; j++) {
            float vj = vals[j];
            r += (vj > vi) || (vj == vi && j < tid);
        }
        if (r < KTOP) sel[r] = tid;
    }
    __syncthreads();

    // gather top-30 rows
    for (int idx = tid; idx < KTOP * 97; idx += 256) {
        int r = idx / 97, c = idx - r * 97;
        pooled[r * 98 + c] = hcat[(g * NPG + sel[r]) * HSTRIDE + c];
    }
    __syncthreads();

    // conv5: kernel 97 / stride 97 over [1, 30*97]  == pooled[30x97] @ w5^T
    for (int idx = tid; idx < KTOP * 16; idx += 256) {
        int t = idx / 16, co = idx - t * 16;
        float s = b5[co];
        const float* pr = &pooled[t * 98];
        const float* wr = &w5[co * 97];
        for (int i = 0; i < 97; i++) s += pr[i] * wr[i];
        A5[t * 16 + co] = fmaxf(s, 0.0f);
    }
    __syncthreads();

    // maxpool(2,2): [30] -> [15] positions
    for (int idx = tid; idx < 15 * 16; idx += 256) {
        int t = idx / 16, co = idx - t * 16;
        P[t * 16 + co] = fmaxf(A5[(2 * t) * 16 + co], A5[(2 * t + 1) * 16 + co]);
    }
    __syncthreads();

    // conv6: 16ch x k5 -> 32ch, 11 positions
    for (int idx = tid; idx < 352; idx += 256) {
        int co = idx / 11, t = idx - co * 11;
        float s = b6[co];
#pragma unroll
        for (int ci = 0; ci < 16; ci++) {
#pragma unroll
            for (int k = 0; k < 5; k++)
                s += w6[(co * 16 + ci) * 5 + k] * P[(t + k) * 16 + ci];
        }
        Z[co * 11 + t] = fmaxf(s, 0.0f);
    }
    __syncthreads();

    // fc1: 352 -> 128, relu
    if (tid < 128) {
        float s = f1b[tid];
        for (int i = 0; i < 352; i++) s += Z[i] * f1w[i * 128 + tid];
        O1[tid] = fmaxf(s, 0.0f);
    }
    __syncthreads();

    // fc2: 128 -> 10
    if (tid < 10) {
        float s = f2b[tid];
#pragma unroll
        for (int j = 0; j < 128; j++) s += O1[j] * f2w[j * 10 + tid];
        O2[tid] = s;
    }
    __syncthreads();

    if (tid == 0) {
        float m = O2[0];
        for (int c = 1; c < 10; c++) m = fmaxf(m, O2[c]);
        float acc = 0.0f;
        for (int c = 0; c < 10; c++) acc += expf(O2[c] - m);
        red[0] = m;
        red[1] = logf(acc);
    }
    __syncthreads();

    if (tid < 10) out[g * 10 + tid] = O2[tid] - red[0] - red[1];
}

// ---------------------------------------------------------------------------
extern "C" void kernel_launch(void* const* d_in, const int* in_sizes, int n_in,
                              void* d_out, int out_size, void* d_ws,
                              size_t ws_size, hipStream_t stream) {
    const float* x   = (const float*)d_in[0];
    const int* ei    = (const int*)d_in[1];
    // d_in[2] = batch (implied by node index, unused)
    const float* W1 = (const float*)d_in[3];
    const float* b1 = (const float*)d_in[4];
    const float* W2 = (const float*)d_in[5];
    const float* b2 = (const float*)d_in[6];
    const float* W3 = (const float*)d_in[7];
    const float* b3 = (const float*)d_in[8];
    const float* W4 = (const float*)d_in[9];
    const float* b4 = (const float*)d_in[10];
    const float* c5w = (const float*)d_in[11];
    const float* c5b = (const float*)d_in[12];
    const float* c6w = (const float*)d_in[13];
    const float* c6b = (const float*)d_in[14];
    const float* f1w = (const float*)d_in[15];
    const float* f1b = (const float*)d_in[16];
    const float* f2w = (const float*)d_in[17];
    const float* f2b = (const float*)d_in[18];
    float* out = (float*)d_out;

    // workspace carve-up (≈54 MB)
    char* ws = (char*)d_ws;
    size_t off = 0;
    auto carve = [&](size_t bytes) {
        void* p = ws + off;
        off = (off + bytes + 255) & ~(size_t)255;
        return p;
    };
    int*      deg    = (int*)carve(N_NODES * sizeof(int));
    float*    dis    = (float*)carve(N_NODES * sizeof(float));
    int*      gcnt   = (int*)carve(N_GRAPHS * sizeof(int));
    int*      goffs  = (int*)carve(N_GRAPHS * sizeof(int));
    int*      gcur   = (int*)carve(N_GRAPHS * sizeof(int));
    uint32_t* binned = (uint32_t*)carve((size_t)N_EDGES * sizeof(uint32_t));
    float*    hcat   = (float*)carve((size_t)N_NODES * HSTRIDE * sizeof(float));
    (void)ws_size; (void)in_sizes; (void)n_in; (void)out_size;

    const int gN = (N_NODES + 255) / 256;
    const int gE = (N_EDGES + 255) / 256;

    k_zero<<<gN, 256, 0, stream>>>(deg, gcnt, gcur);
    k_deg<<<gE, 256, 0, stream>>>(ei, deg, gcnt);
    k_rsqrt<<<gN, 256, 0, stream>>>(deg, dis);
    k_scan<<<1, 1024, 0, stream>>>(gcnt, goffs);
    k_scatter<<<gE, 256, 0, stream>>>(ei, goffs, gcur, binned);

    // GCN stack: concat columns [x1:0..31, x2:32..63, x3:64..95, x4:96]
    k_gcn<128, 32><<<N_GRAPHS, 256, 0, stream>>>(x, 128, 0, W1, b1, dis, binned,
                                                 goffs, gcnt, hcat, 0);
    k_gcn<32, 32><<<N_GRAPHS, 256, 0, stream>>>(hcat, HSTRIDE, 0, W2, b2, dis,
                                                binned, goffs, gcnt, hcat, 32);
    k_gcn<32, 32><<<N_GRAPHS, 256, 0, stream>>>(hcat, HSTRIDE, 32, W3, b3, dis,
                                                binned, goffs, gcnt, hcat, 64);
    k_gcn<32, 1><<<N_GRAPHS, 256, 0, stream>>>(hcat, HSTRIDE, 64, W4, b4, dis,
                                               binned, goffs, gcnt, hcat, 96);

    k_head<<<N_GRAPHS, 256, 0, stream>>>(hcat, c5w, c5b, c6w, c6b, f1w, f1b,
                                         f2w, f2b, out);
}